// BaseRVQ_53618371723759
// MI455X (gfx1250) — compile-verified
//
#include <hip/hip_runtime.h>

typedef float v2f __attribute__((ext_vector_type(2)));
typedef float v8f __attribute__((ext_vector_type(8)));

#define N_ROWS   65536
#define DDIM     256
#define KCODES   512
#define NQ       4
#define ROWS_WG  128
#define THREADS  256
#define LSTRIDE  260              // 256 + 4 pad -> conflict-free b64 column reads
#define CHUNK    32               // codes per LDS buffer (double-buffered)
#define NCHUNK   (KCODES / CHUNK) // 16
#define NUM_WG   (N_ROWS / ROWS_WG) // 512
#define CPY4     (CHUNK * DDIM / 4 / THREADS)  // float4 per thread per chunk = 8

// ---------------------------------------------------------------------------
// Kernel 1: half squared norms  cnorm[q*512+k] = 0.5 * sum_d c[q][k][d]^2
// ---------------------------------------------------------------------------
__global__ void rvq_cnorm(const float* __restrict__ cb, float* __restrict__ cnorm) {
    int row = blockIdx.x * blockDim.x + threadIdx.x;   // 0..2047
    if (row < NQ * KCODES) {
        const float* p = cb + (size_t)row * DDIM;
        float s = 0.f;
        for (int d = 0; d < DDIM; ++d) s += p[d] * p[d];
        cnorm[row] = 0.5f * s;
    }
}

// ---------------------------------------------------------------------------
// Kernel 2: main RVQ. 512 WGs x 256 threads (8 waves), 128 rows per WG.
// Scores via V_WMMA_F32_16X16X4_F32: acc = 0.5*||c||^2 - r.c  (0.5x true dist)
// Codebook chunks double-buffered in LDS; next chunk register-staged so the
// global-load latency hides under the WMMA stream; one barrier per chunk.
// ---------------------------------------------------------------------------
__global__ void rvq_main(const float* __restrict__ x,        // encoded_x [16,256,1024,4]
                         const float* __restrict__ cb,       // codebooks [4,512,256]
                         const float* __restrict__ cnorm,    // [4*512] (pre-halved)
                         float* __restrict__ partials,       // [4*NUM_WG] loss partials
                         float* __restrict__ out,            // [N,256]
                         int*   __restrict__ idx_out) {      // [N,4]
    extern __shared__ float smem[];
    float* resid  = smem;                                     // 128 x 260
    float* cbuf0  = smem + ROWS_WG * LSTRIDE;                 // 2 x (32 x 260), negated
    int*   wgIdx  = (int*)(cbuf0 + 2 * CHUNK * LSTRIDE);      // 128
    float* redBuf = (float*)(wgIdx + ROWS_WG);                // 256

    const int tid = threadIdx.x;
    const int wg  = blockIdx.x;
    const int n0  = wg * ROWS_WG;
    // x_in[n,d] = encoded_x[b, d, h, w]; n = ((b*4 + w)*1024 + h)
    const int b  = n0 >> 12;
    const int w  = (n0 >> 10) & 3;
    const int h0 = n0 & 1023;

    // ---- gather rows into LDS residual (thread = (row, half-of-D)) ----
    {
        const int r = tid >> 1;
        const int dbase = (tid & 1) * 128;
        const float* src = x + (size_t)b * 1048576 + (size_t)(h0 + r) * 4 + w;
        float* dst = resid + r * LSTRIDE;
        for (int j = 0; j < 128; ++j) {
            int d = dbase + j;
            dst[d] = src[(size_t)d * 4096];
        }
    }

    const int lane  = tid & 31;
    const int wave  = tid >> 5;
    const int half  = lane >> 4;   // selects K-pair / C-row group
    const int lr    = lane & 15;
    const int m0    = wave * 16;   // 16 rows per wave (8 waves -> 128 rows)

    for (int q = 0; q < NQ; ++q) {
        const float* cbq = cb + (size_t)q * KCODES * DDIM;

        // ---- prologue: stage chunk 0 (negated) into buffer 0 ----
        {
            const float4* s4 = (const float4*)cbq;
#pragma unroll
            for (int j = 0; j < CPY4; ++j) {
                int i = tid + j * THREADS;          // 0..2047
                int row = i >> 6, d4 = i & 63;
                float4 v4 = s4[row * 64 + d4];
                float4 nv; nv.x = -v4.x; nv.y = -v4.y; nv.z = -v4.z; nv.w = -v4.w;
                *(float4*)(cbuf0 + row * LSTRIDE + d4 * 4) = nv;
            }
        }
        __syncthreads();   // buf0 ready; also fences resid writes (init / prev stage)

        float bestS[8];
        int   bestI[8];
#pragma unroll
        for (int v = 0; v < 8; ++v) { bestS[v] = 3.4e38f; bestI[v] = 0; }

        for (int kc = 0; kc < NCHUNK; ++kc) {
            // ---- (1) issue global loads for next chunk into registers ----
            float4 tmp[CPY4];
            if (kc + 1 < NCHUNK) {
                const float4* s4 = (const float4*)(cbq + (size_t)(kc + 1) * CHUNK * DDIM);
#pragma unroll
                for (int j = 0; j < CPY4; ++j) {
                    int i = tid + j * THREADS;
                    tmp[j] = s4[(i >> 6) * 64 + (i & 63)];
                }
            }

            // ---- (2) 16x32 score tile per wave from current buffer ----
            const float* cur = cbuf0 + (kc & 1) * (CHUNK * LSTRIDE);
            v8f acc[2];
#pragma unroll
            for (int t = 0; t < 2; ++t) {
                float cn = cnorm[q * KCODES + kc * CHUNK + t * 16 + lr]; // 0.5*||c||^2
#pragma unroll
                for (int v = 0; v < 8; ++v) acc[t][v] = cn;
            }
            const float* aptr = resid + (m0 + lr) * LSTRIDE + 2 * half;
            const float* bptr = cur + lr * LSTRIDE + 2 * half;
#pragma unroll 8
            for (int d0 = 0; d0 < DDIM; d0 += 4) {
                v2f a  = *(const v2f*)(aptr + d0);
                v2f b0 = *(const v2f*)(bptr + d0);
                v2f b1 = *(const v2f*)(bptr + 16 * LSTRIDE + d0);
                acc[0] = __builtin_amdgcn_wmma_f32_16x16x4_f32(
                    false, a, false, b0, (short)0, acc[0], false, false);
                acc[1] = __builtin_amdgcn_wmma_f32_16x16x4_f32(
                    false, a, false, b1, (short)0, acc[1], false, false);
            }

            // ---- (3) write staged chunk (negated) into the other buffer ----
            if (kc + 1 < NCHUNK) {
                float* nxt = cbuf0 + ((kc + 1) & 1) * (CHUNK * LSTRIDE);
#pragma unroll
                for (int j = 0; j < CPY4; ++j) {
                    int i = tid + j * THREADS;
                    int row = i >> 6, d4 = i & 63;
                    float4 nv;
                    nv.x = -tmp[j].x; nv.y = -tmp[j].y; nv.z = -tmp[j].z; nv.w = -tmp[j].w;
                    *(float4*)(nxt + row * LSTRIDE + d4 * 4) = nv;
                }
            }

            // ---- lane-local running argmin (codes ascend => strict <) ----
#pragma unroll
            for (int t = 0; t < 2; ++t) {
                int kg = kc * CHUNK + t * 16 + lr;
#pragma unroll
                for (int v = 0; v < 8; ++v) {
                    float s = acc[t][v];
                    if (s < bestS[v]) { bestS[v] = s; bestI[v] = kg; }
                }
            }

            __syncthreads();   // next buffer visible; everyone done with current
        } // kc

        // ---- cross-lane argmin within each 16-lane half group ----
#pragma unroll
        for (int v = 0; v < 8; ++v) {
            float s = bestS[v]; int i = bestI[v];
            for (int m = 8; m >= 1; m >>= 1) {
                float os = __shfl_xor(s, m, 32);
                int   oi = __shfl_xor(i, m, 32);
                if (os < s || (os == s && oi < i)) { s = os; i = oi; }
            }
            if (lr == 0) wgIdx[m0 + v + 8 * half] = i;   // row = m0 + v + 8*half
        }
        __syncthreads();

        // ---- residual update, loss partial, index write ----
        float sq = 0.f;
        {
            const int r = tid >> 1;
            const int dbase = (tid & 1) * 128;
            const int ci = wgIdx[r];
            const float4* crow = (const float4*)(cbq + (size_t)ci * DDIM + dbase);
            float* rrow = resid + r * LSTRIDE + dbase;
            for (int j = 0; j < 32; ++j) {
                float4 c4 = crow[j];
                float4 r4 = *(float4*)(rrow + j * 4);
                r4.x -= c4.x; r4.y -= c4.y; r4.z -= c4.z; r4.w -= c4.w;
                *(float4*)(rrow + j * 4) = r4;
                sq += r4.x * r4.x + r4.y * r4.y + r4.z * r4.z + r4.w * r4.w;
            }
            if ((tid & 1) == 0) idx_out[(size_t)(n0 + r) * NQ + q] = ci;
        }
        redBuf[tid] = sq;
        __syncthreads();
        for (int s2 = THREADS / 2; s2 > 0; s2 >>= 1) {
            if (tid < s2) redBuf[tid] += redBuf[tid + s2];
            __syncthreads();
        }
        if (tid == 0) partials[q * NUM_WG + wg] = redBuf[0];
        __syncthreads();
    } // q

    // ---- quantized_out = x_in - residual_final ----
    {
        const int r = tid >> 1;
        const int dbase = (tid & 1) * 128;
        const float* src = x + (size_t)b * 1048576 + (size_t)(h0 + r) * 4 + w;
        float* orow = out + (size_t)(n0 + r) * DDIM;
        const float* rrow = resid + r * LSTRIDE;
        for (int j = 0; j < 32; ++j) {
            int d = dbase + j * 4;
            float4 o;
            o.x = src[(size_t)(d + 0) * 4096] - rrow[d + 0];
            o.y = src[(size_t)(d + 1) * 4096] - rrow[d + 1];
            o.z = src[(size_t)(d + 2) * 4096] - rrow[d + 2];
            o.w = src[(size_t)(d + 3) * 4096] - rrow[d + 3];
            *(float4*)(orow + d) = o;
        }
    }
}

// ---------------------------------------------------------------------------
// Kernel 3: deterministic fixed-order loss reduction
// ---------------------------------------------------------------------------
__global__ void rvq_loss(const float* __restrict__ partials, float* __restrict__ loss_out) {
    __shared__ float buf[256];
    const int q = blockIdx.x;
    float s = 0.f;
    for (int i = threadIdx.x; i < NUM_WG; i += 256) s += partials[q * NUM_WG + i];
    buf[threadIdx.x] = s;
    __syncthreads();
    for (int st = 128; st > 0; st >>= 1) {
        if (threadIdx.x < st) buf[threadIdx.x] += buf[threadIdx.x + st];
        __syncthreads();
    }
    if (threadIdx.x == 0)
        loss_out[q] = buf[0] / (float)((size_t)N_ROWS * DDIM);
}

extern "C" void kernel_launch(void* const* d_in, const int* in_sizes, int n_in,
                              void* d_out, int out_size, void* d_ws, size_t ws_size,
                              hipStream_t stream) {
    const float* x  = (const float*)d_in[0];   // encoded_x [16,256,1024,4]
    const float* cb = (const float*)d_in[1];   // codebooks [4,512,256]

    float* cnorm    = (float*)d_ws;                    // 2048 floats
    float* partials = cnorm + NQ * KCODES;             // 2048 floats

    float* out      = (float*)d_out;                           // [N*256] f32
    int*   idx_out  = (int*)(out + (size_t)N_ROWS * DDIM);     // [N*4] i32
    float* loss_out = (float*)(idx_out + (size_t)N_ROWS * NQ); // [4] f32

    rvq_cnorm<<<8, 256, 0, stream>>>(cb, cnorm);

    size_t lds_bytes = (size_t)(ROWS_WG * LSTRIDE + 2 * CHUNK * LSTRIDE
                                + ROWS_WG + THREADS) * sizeof(float);
    rvq_main<<<NUM_WG, THREADS, lds_bytes, stream>>>(x, cb, cnorm, partials, out, idx_out);

    rvq_loss<<<NQ, 256, 0, stream>>>(partials, loss_out);
}